// DWLMLayer_82961588289635
// MI455X (gfx1250) — compile-verified
//
#include <hip/hip_runtime.h>
#include <hip/hip_bf16.h>

// Problem constants (from the reference)
constexpr int A_TOTAL     = 5456;   // 4096+1024+256+64+16
constexpr int BATCH       = 16;
constexpr int LEVELS      = 5;
constexpr int MAX_OBJ     = 32;
constexpr int NUM_CLASSES = 80;
constexpr int TOP_K       = 3;
constexpr int CHUNKS      = 171;    // ceil(5456/32); boundaries 4096/5120/5376/5440 all %32==0

typedef __attribute__((ext_vector_type(2))) float v2f;
typedef __attribute__((ext_vector_type(8))) float v8f;

__device__ __forceinline__ int level_of_chunk(int c) {
    return (c < 128) ? 0 : (c < 160) ? 1 : (c < 168) ? 2 : (c < 170) ? 3 : 4;
}
__device__ __forceinline__ int level_of_anchor(int a) {
    return (a < 4096) ? 0 : (a < 5120) ? 1 : (a < 5376) ? 2 : (a < 5440) ? 3 : 4;
}

// ---------------------------------------------------------------- kernel 0
__global__ void dwlm_zero_kernel(float* __restrict__ ws, int n) {
    int i = blockIdx.x * blockDim.x + threadIdx.x;
    if (i < n) ws[i] = 0.0f;
}

// ---------------------------------------------------------------- kernel 1
// Fused focal + GIoU loss, then WMMA-based binning of
// einsum('ba,bola->bol') into per-(b,level) 32-object sum/count bins.
__global__ __launch_bounds__(256)
void dwlm_loss_bin_kernel(const float* __restrict__ cls_pred,   // [B,A,80]
                          const float* __restrict__ loc_pred,   // [B,A,4]
                          const float* __restrict__ cls_tar,    // [B,A,82]
                          const float* __restrict__ loc_tar,    // [B,A,4]
                          const int*   __restrict__ ind_tar,    // [B,A,1]
                          float* __restrict__ sum_ws,           // [B,L,32]
                          float* __restrict__ cnt_ws,           // [B,L,32]
                          float* __restrict__ pos_out)          // [B*A]
{
    const int lane   = threadIdx.x & 31;
    const int waveId = blockIdx.x * (blockDim.x >> 5) + (threadIdx.x >> 5);
    const int b      = waveId / CHUNKS;            // grid is exactly B*CHUNKS waves
    const int chunk  = waveId - b * CHUNKS;
    const int a      = chunk * 32 + lane;
    const bool valid = (a < A_TOTAL);
    const int  lvl   = level_of_chunk(chunk);

    float loss = 0.0f;
    int   ind  = 0;

    if (valid) {
        const int row = b * A_TOTAL + a;

        // ---- focal loss over 80 classes ----
        const float* pp = cls_pred + (size_t)row * NUM_CLASSES;
        const float* tt = cls_tar  + (size_t)row * (NUM_CLASSES + 2);
        float fsum = 0.0f;
        #pragma unroll 4
        for (int c = 0; c < NUM_CLASSES; ++c) {
            float p = fminf(fmaxf(pp[c], 1e-7f), 1.0f - 1e-7f);
            float y = tt[c];
            float ce      = -(y * __logf(p) + (1.0f - y) * __logf(1.0f - p));
            float alpha_t = y * 0.25f + (1.0f - y) * 0.75f;
            float pt      = y * (1.0f - p) + (1.0f - y) * p;
            fsum += alpha_t * pt * pt * ce;
        }
        pos_out[row] = tt[NUM_CLASSES + 1];  // pos_mask = cls_tar[..., -1]

        // ---- GIoU loss (ltrb distances) ----
        const float4 t4 = *(const float4*)(loc_tar  + (size_t)row * 4);
        const float4 p4 = *(const float4*)(loc_pred + (size_t)row * 4);
        float tl = t4.x, tth = t4.y, tr = t4.z, tb = t4.w;
        float pl = p4.x, pth = p4.y, pr = p4.z, pb = p4.w;
        float area_t = (tl + tr) * (tth + tb);
        float area_p = (pl + pr) * (pth + pb);
        float inter  = (fminf(tl, pl) + fminf(tr, pr)) * (fminf(tth, pth) + fminf(tb, pb));
        float uni    = area_t + area_p - inter;
        float iou    = inter / fmaxf(uni, 1e-7f);
        float enc    = (fmaxf(tl, pl) + fmaxf(tr, pr)) * (fmaxf(tth, pth) + fmaxf(tb, pb));
        float giou   = iou - (enc - uni) / fmaxf(enc, 1e-7f);

        loss = fsum + (1.0f - giou);
        ind  = ind_tar[row];
    }

    // ---- WMMA binning: D[m,n] = sum_k A[m,k] * B[k,n] over K=4 anchors ----
    // A[m,k] = ((ind_k & 15) == m) && valid_k            (one-hot, f32)
    // B[k,n] = loss_k  if n == (ind_k>>4)                (columns 0,1: sums lo/hi)
    //        = 1.0     if n == 2 + (ind_k>>4)            (columns 2,3: counts lo/hi)
    // ISA layout: A/B lane L, VGPR j  <-> K = 2*(L>=16) + j ; M/N = L & 15
    const float vflag = valid ? 1.0f : 0.0f;
    const int   half  = lane >> 4;     // 0/1
    const int   mn    = lane & 15;     // both A's M and B's N for this lane

    v8f acc = {0.0f, 0.0f, 0.0f, 0.0f, 0.0f, 0.0f, 0.0f, 0.0f};

    #pragma unroll
    for (int g = 0; g < 8; ++g) {
        const int src = g * 4 + half * 2;             // anchor lane for VGPR j=0
        float l0 = __shfl(loss,  src,     32);
        float l1 = __shfl(loss,  src + 1, 32);
        int   i0 = __shfl(ind,   src,     32);
        int   i1 = __shfl(ind,   src + 1, 32);
        float v0 = __shfl(vflag, src,     32);
        float v1 = __shfl(vflag, src + 1, 32);

        v2f Am, Bm;
        Am.x = ((i0 & 15) == mn) ? v0 : 0.0f;
        Am.y = ((i1 & 15) == mn) ? v1 : 0.0f;
        const int hi0 = (i0 >> 4) & 1;
        const int hi1 = (i1 >> 4) & 1;
        Bm.x = (mn == hi0) ? l0 : ((mn == 2 + hi0) ? 1.0f : 0.0f);
        Bm.y = (mn == hi1) ? l1 : ((mn == 2 + hi1) ? 1.0f : 0.0f);

        acc = __builtin_amdgcn_wmma_f32_16x16x4_f32(
            /*neg_a=*/false, Am, /*neg_b=*/false, Bm,
            /*c_mod=*/(short)0, acc, /*reuse_a=*/false, /*reuse_b=*/false);
    }

    // ---- flush: D lane L, VGPR r <-> M = r + 8*half, N = mn ----
    if (mn < 4) {
        float* dst = (mn < 2) ? sum_ws : cnt_ws;          // N 0,1 = sums; 2,3 = counts
        float* seg = dst + (size_t)(b * LEVELS + lvl) * MAX_OBJ;
        const int obj_base = ((mn & 1) << 4) + (half << 3);  // 16*(n&1) + 8*half
        #pragma unroll
        for (int r = 0; r < 8; ++r) {
            atomicAdd(seg + obj_base + r, acc[r]);
        }
    }
}

// ---------------------------------------------------------------- kernel 2
// Per (b, o): mean over levels, min/max normalization, top-3 threshold, validity.
__global__ void dwlm_target_kernel(const float* __restrict__ sum_ws,
                                   const float* __restrict__ cnt_ws,
                                   const int*   __restrict__ bbox_cnt,   // [B,1]
                                   float* __restrict__ tgt_ws)           // [B,L,32]
{
    int t = blockIdx.x * blockDim.x + threadIdx.x;
    if (t >= BATCH * MAX_OBJ) return;
    const int b = t / MAX_OBJ;
    const int o = t - b * MAX_OBJ;

    float mean[LEVELS];
    float lmax = -1e30f;
    #pragma unroll
    for (int l = 0; l < LEVELS; ++l) {
        const int idx = (b * LEVELS + l) * MAX_OBJ + o;
        mean[l] = sum_ws[idx] / fmaxf(1.0f, cnt_ws[idx]);
        lmax = fmaxf(lmax, mean[l]);
    }
    lmax += 1e-5f;
    float lmin = 1e30f;
    #pragma unroll
    for (int l = 0; l < LEVELS; ++l) {
        if (mean[l] == 0.0f) mean[l] = lmax;
        lmin = fminf(lmin, mean[l]);
    }
    float tv[LEVELS], srt[LEVELS];
    #pragma unroll
    for (int l = 0; l < LEVELS; ++l) {
        tv[l]  = 1.0f - (mean[l] - lmin) / (lmax - lmin);
        srt[l] = tv[l];
    }
    // descending bubble sort of 5 values -> kth = 3rd largest
    #pragma unroll
    for (int i = 0; i < LEVELS; ++i)
        #pragma unroll
        for (int j = 0; j < LEVELS - 1 - i; ++j)
            if (srt[j] < srt[j + 1]) { float tmp = srt[j]; srt[j] = srt[j + 1]; srt[j + 1] = tmp; }
    const float kth   = srt[TOP_K - 1];
    const bool  validO = (o < bbox_cnt[b]);

    #pragma unroll
    for (int l = 0; l < LEVELS; ++l) {
        float v = (tv[l] >= kth) ? tv[l] : 0.0f;
        if (!validO) v = 0.0f;
        tgt_ws[(b * LEVELS + l) * MAX_OBJ + o] = v;
    }
}

// ---------------------------------------------------------------- kernel 3
// Scatter back: out[b,a] = pos_mask>0 ? target[b, ind[b,a], level(a)] : 1.0
__global__ void dwlm_gather_kernel(const int*   __restrict__ ind_tar,
                                   const float* __restrict__ tgt_ws,
                                   const float* __restrict__ pos_mask,
                                   float* __restrict__ out)
{
    int i = blockIdx.x * blockDim.x + threadIdx.x;
    if (i >= BATCH * A_TOTAL) return;
    const int b   = i / A_TOTAL;
    const int a   = i - b * A_TOTAL;
    const int lvl = level_of_anchor(a);
    const int o   = ind_tar[i];
    const float t = tgt_ws[(b * LEVELS + lvl) * MAX_OBJ + o];
    out[i] = (pos_mask[i] > 0.0f) ? t : 1.0f;
}

// ---------------------------------------------------------------- launcher
extern "C" void kernel_launch(void* const* d_in, const int* in_sizes, int n_in,
                              void* d_out, int out_size, void* d_ws, size_t ws_size,
                              hipStream_t stream) {
    const float* cls_pred = (const float*)d_in[0];
    const float* loc_pred = (const float*)d_in[1];
    const float* cls_tar  = (const float*)d_in[2];
    const float* loc_tar  = (const float*)d_in[3];
    const int*   ind_tar  = (const int*)d_in[4];
    const int*   bbox_cnt = (const int*)d_in[5];

    float* out = (float*)d_out;                     // [B*A] dwlm_out_masked
    float* pos = out + BATCH * A_TOTAL;             // [B*A] pos_mask output

    float* sum_ws = (float*)d_ws;                   // [B*L*32]
    float* cnt_ws = sum_ws + BATCH * LEVELS * MAX_OBJ;
    float* tgt_ws = cnt_ws + BATCH * LEVELS * MAX_OBJ;

    // 0) zero the accumulator bins (ws is poisoned by the harness)
    const int nz = 2 * BATCH * LEVELS * MAX_OBJ;
    dwlm_zero_kernel<<<(nz + 255) / 256, 256, 0, stream>>>(sum_ws, nz);

    // 1) fused loss + WMMA binning: exactly B*CHUNKS waves, 8 waves/block
    const int nwaves = BATCH * CHUNKS;              // 2736 -> 342 blocks
    dwlm_loss_bin_kernel<<<nwaves / 8, 256, 0, stream>>>(
        cls_pred, loc_pred, cls_tar, loc_tar, ind_tar, sum_ws, cnt_ws, pos);

    // 2) per-(b,o) target computation
    dwlm_target_kernel<<<(BATCH * MAX_OBJ + 255) / 256, 256, 0, stream>>>(
        sum_ws, cnt_ws, bbox_cnt, tgt_ws);

    // 3) gather + mask
    const int n = BATCH * A_TOTAL;
    dwlm_gather_kernel<<<(n + 255) / 256, 256, 0, stream>>>(ind_tar, tgt_ws, pos, out);
}